// GKT_29961691857540
// MI455X (gfx1250) — compile-verified
//
#include <hip/hip_runtime.h>
#include <hip/hip_bf16.h>
#include <math.h>

#define B_ 32
#define T_ 50
#define C_ 501
#define H_ 32
#define E_ 32
#define BN_SCALE_ 0.99999500003749969f /* 1/sqrt(1+1e-5) */

typedef __attribute__((ext_vector_type(2))) float v2f;
typedef __attribute__((ext_vector_type(8))) float v8f;

__device__ __forceinline__ v8f z8() {
    v8f z = {0.f, 0.f, 0.f, 0.f, 0.f, 0.f, 0.f, 0.f};
    return z;
}

// D(16x16 f32) = A(16x4 f32) x B(4x16 f32) + C   -- full fp32 precision
__device__ __forceinline__ v8f wmma4(v2f a, v2f b, v8f c) {
    return __builtin_amdgcn_wmma_f32_16x16x4_f32(false, a, false, b, (short)0, c,
                                                 false, false);
}

__device__ __forceinline__ float sigm(float x) { return 1.0f / (1.0f + __expf(-x)); }

// One recurrent timestep: h_next(B,C,H) from ht(B,C,H).
// Grid: B*8 blocks (8 tiles of 64 concepts), 128 threads (4 waves x 16-node tiles).
__global__ __launch_bounds__(128) void gkt_step(
    const int* __restrict__ features, const int* __restrict__ questions,
    const float* __restrict__ graph, const float* __restrict__ emb_x,
    const float* __restrict__ emb_c,
    const float* __restrict__ self_w1, const float* __restrict__ self_b1,
    const float* __restrict__ self_w2, const float* __restrict__ self_b2,
    const float* __restrict__ n0_w1, const float* __restrict__ n0_b1,
    const float* __restrict__ n0_w2, const float* __restrict__ n0_b2,
    const float* __restrict__ n1_w1, const float* __restrict__ n1_b1,
    const float* __restrict__ n1_w2, const float* __restrict__ n1_b2,
    const float* __restrict__ ea_w, const float* __restrict__ ea_ew,
    const float* __restrict__ ea_eb, const float* __restrict__ ea_aw,
    const float* __restrict__ ea_ab,
    const float* __restrict__ w_ih, const float* __restrict__ w_hh,
    const float* __restrict__ b_ih, const float* __restrict__ b_hh,
    const float* __restrict__ ht_in, float* __restrict__ ht_out, int t)
{
    __shared__ float selfvecS[64];           // [ht[b,q,:], emb_x[xt,:]]
    __shared__ float h1S[32];
    __shared__ float selffS[32];             // self MLP output
    __shared__ float adjS[64], radjS[64], wS[64];
    __shared__ float stage[4][2][16][36];    // per-wave D->A staging tiles

    const int tid  = threadIdx.x;
    const int wv   = tid >> 5;
    const int lane = tid & 31;
    const int hi   = lane >> 4;              // half-wave select
    const int mrow = lane & 15;              // M row / N col within 16
    const int b      = blockIdx.x >> 3;
    const int cbaseB = (blockIdx.x & 7) * 64;
    const int cbase  = cbaseB + wv * 16;

    const int q  = questions[b * T_ + t];
    const int xt = features[b * T_ + t];

    // ---- block preamble: stage per-row scalars + self vector ----
    if (tid < 64) {
        int c  = cbaseB + tid;
        int cc = c < C_ ? c : C_ - 1;
        adjS[tid]  = graph[q * C_ + cc];
        radjS[tid] = graph[cc * C_ + q];
        wS[tid]    = ea_w[cc];
    } else {
        int k = tid - 64;
        selfvecS[k] = (k < H_) ? ht_in[((size_t)b * C_ + q) * H_ + k]
                               : emb_x[xt * E_ + (k - H_)];
    }
    __syncthreads();
    if (tid < 32) {
        float s = self_b1[tid];
#pragma unroll
        for (int k = 0; k < 64; ++k) s += self_w1[tid * 64 + k] * selfvecS[k];
        h1S[tid] = fmaxf(s, 0.f);
    }
    __syncthreads();
    if (tid < 32) {
        float s = self_b2[tid];
#pragma unroll
        for (int k = 0; k < 32; ++k) s += self_w2[tid * 32 + k] * h1S[k];
        selffS[tid] = fmaxf(s, 0.f) * BN_SCALE_;
    }
    __syncthreads();

    const int cN  = cbase + mrow;
    const int ccN = cN < C_ ? cN : C_ - 1;

    // ---- A-fragments for ht[b,c,:] (reused for layer-1 K=64..95 and GRU gh) ----
    v2f aht[8];
#pragma unroll
    for (int ks = 0; ks < 8; ++ks) {
        int k0 = 4 * ks + 2 * hi;
        const float* p = &ht_in[((size_t)b * C_ + ccN) * H_ + k0];
        aht[ks].x = p[0];
        aht[ks].y = p[1];
    }
    // ---- A-fragments for ce[b,c,:] (K=96..127) ----
    v2f ace[8];
    const float* cerow = (ccN == q) ? (emb_x + (size_t)xt * E_) : (emb_c + (size_t)ccN * E_);
#pragma unroll
    for (int ks = 0; ks < 8; ++ks) {
        int k0 = 4 * ks + 2 * hi;
        ace[ks].x = cerow[k0];
        ace[ks].y = cerow[k0 + 1];
    }

    // ---- layer 1: (16x128) @ (128x32)^T for n0 and n1, shared A ----
    v8f acc00 = z8(), acc01 = z8(), acc10 = z8(), acc11 = z8();
#pragma unroll
    for (int ks = 0; ks < 32; ++ks) {
        int k0 = 4 * ks + 2 * hi;
        v2f a;
        if (ks < 16) { a.x = selfvecS[k0]; a.y = selfvecS[k0 + 1]; }
        else if (ks < 24) a = aht[ks - 16];
        else              a = ace[ks - 24];
        v2f b00, b01, b10, b11;
        b00.x = n0_w1[(mrow)      * 128 + k0]; b00.y = n0_w1[(mrow)      * 128 + k0 + 1];
        b01.x = n0_w1[(mrow + 16) * 128 + k0]; b01.y = n0_w1[(mrow + 16) * 128 + k0 + 1];
        b10.x = n1_w1[(mrow)      * 128 + k0]; b10.y = n1_w1[(mrow)      * 128 + k0 + 1];
        b11.x = n1_w1[(mrow + 16) * 128 + k0]; b11.y = n1_w1[(mrow + 16) * 128 + k0 + 1];
        acc00 = wmma4(a, b00, acc00);
        acc01 = wmma4(a, b01, acc01);
        acc10 = wmma4(a, b10, acc10);
        acc11 = wmma4(a, b11, acc11);
    }
    {
        float bn0a = n0_b1[mrow], bn0b = n0_b1[mrow + 16];
        float bn1a = n1_b1[mrow], bn1b = n1_b1[mrow + 16];
#pragma unroll
        for (int i = 0; i < 8; ++i) {
            acc00[i] = fmaxf(acc00[i] + bn0a, 0.f);
            acc01[i] = fmaxf(acc01[i] + bn0b, 0.f);
            acc10[i] = fmaxf(acc10[i] + bn1a, 0.f);
            acc11[i] = fmaxf(acc11[i] + bn1b, 0.f);
        }
    }
    // D-layout -> LDS (per-wave tiles; same-wave DS is in-order)
#pragma unroll
    for (int i = 0; i < 8; ++i) {
        int r = i + 8 * hi;
        stage[wv][0][r][mrow]      = acc00[i];
        stage[wv][0][r][mrow + 16] = acc01[i];
        stage[wv][1][r][mrow]      = acc10[i];
        stage[wv][1][r][mrow + 16] = acc11[i];
    }
    __builtin_amdgcn_wave_barrier();

    // ---- layer 2: (16x32) @ (32x32)^T for n0 and n1 ----
    v8f h0a = z8(), h0b = z8(), h1a = z8(), h1b = z8();
#pragma unroll
    for (int ks = 0; ks < 8; ++ks) {
        int k0 = 4 * ks + 2 * hi;
        v2f a0, a1, b00, b01, b10, b11;
        a0.x = stage[wv][0][mrow][k0]; a0.y = stage[wv][0][mrow][k0 + 1];
        a1.x = stage[wv][1][mrow][k0]; a1.y = stage[wv][1][mrow][k0 + 1];
        b00.x = n0_w2[(mrow)      * 32 + k0]; b00.y = n0_w2[(mrow)      * 32 + k0 + 1];
        b01.x = n0_w2[(mrow + 16) * 32 + k0]; b01.y = n0_w2[(mrow + 16) * 32 + k0 + 1];
        b10.x = n1_w2[(mrow)      * 32 + k0]; b10.y = n1_w2[(mrow)      * 32 + k0 + 1];
        b11.x = n1_w2[(mrow + 16) * 32 + k0]; b11.y = n1_w2[(mrow + 16) * 32 + k0 + 1];
        h0a = wmma4(a0, b00, h0a); h0b = wmma4(a0, b01, h0b);
        h1a = wmma4(a1, b10, h1a); h1b = wmma4(a1, b11, h1b);
    }

    // ---- combine: m = adj*mlp0 + radj*mlp1 ; row q -> self_feat ----
    v8f m0, m1;
    {
        float bb0a = n0_b2[mrow], bb0b = n0_b2[mrow + 16];
        float bb1a = n1_b2[mrow], bb1b = n1_b2[mrow + 16];
#pragma unroll
        for (int i = 0; i < 8; ++i) {
            int r  = i + 8 * hi;
            int li = wv * 16 + r;
            float adj = adjS[li], radj = radjS[li];
            float f0a = fmaxf(h0a[i] + bb0a, 0.f) * BN_SCALE_;
            float f0b = fmaxf(h0b[i] + bb0b, 0.f) * BN_SCALE_;
            float f1a = fmaxf(h1a[i] + bb1a, 0.f) * BN_SCALE_;
            float f1b = fmaxf(h1b[i] + bb1b, 0.f) * BN_SCALE_;
            float ma = adj * f0a + radj * f1a;
            float mb = adj * f0b + radj * f1b;
            if (cbase + r == q) { ma = selffS[mrow]; mb = selffS[mrow + 16]; }
            m0[i] = ma;
            m1[i] = mb;
        }
    }

    // ---- edge attention: eg = sigmoid(m@ea_ew^T+eb), af = tanh(m@ea_aw^T+ab) ----
#pragma unroll
    for (int i = 0; i < 8; ++i) {
        int r = i + 8 * hi;
        stage[wv][0][r][mrow]      = m0[i];
        stage[wv][0][r][mrow + 16] = m1[i];
    }
    __builtin_amdgcn_wave_barrier();
    v2f am[8];
#pragma unroll
    for (int ks = 0; ks < 8; ++ks) {
        int k0 = 4 * ks + 2 * hi;
        am[ks].x = stage[wv][0][mrow][k0];
        am[ks].y = stage[wv][0][mrow][k0 + 1];
    }
    v8f ega = z8(), egb = z8(), afa = z8(), afb = z8();
#pragma unroll
    for (int ks = 0; ks < 8; ++ks) {
        int k0 = 4 * ks + 2 * hi;
        v2f be0, be1, ba0, ba1;
        be0.x = ea_ew[(mrow)      * 32 + k0]; be0.y = ea_ew[(mrow)      * 32 + k0 + 1];
        be1.x = ea_ew[(mrow + 16) * 32 + k0]; be1.y = ea_ew[(mrow + 16) * 32 + k0 + 1];
        ba0.x = ea_aw[(mrow)      * 32 + k0]; ba0.y = ea_aw[(mrow)      * 32 + k0 + 1];
        ba1.x = ea_aw[(mrow + 16) * 32 + k0]; ba1.y = ea_aw[(mrow + 16) * 32 + k0 + 1];
        ega = wmma4(am[ks], be0, ega); egb = wmma4(am[ks], be1, egb);
        afa = wmma4(am[ks], ba0, afa); afb = wmma4(am[ks], ba1, afb);
    }
    {
        float beb0 = ea_eb[mrow], beb1 = ea_eb[mrow + 16];
        float bab0 = ea_ab[mrow], bab1 = ea_ab[mrow + 16];
#pragma unroll
        for (int i = 0; i < 8; ++i) {
            int r  = i + 8 * hi;
            float w = wS[wv * 16 + r];
            float eg0 = sigm(ega[i] + beb0), eg1 = sigm(egb[i] + beb1);
            float af0 = tanhf(afa[i] + bab0), af1 = tanhf(afb[i] + bab1);
            m0[i] = m0[i] - w * eg0 * m0[i] + w * af0;
            m1[i] = m1[i] - w * eg1 * m1[i] + w * af1;
        }
    }

    // ---- GRU: gx = m'@w_ih^T (N=96), gh = ht@w_hh^T (N=96) ----
#pragma unroll
    for (int i = 0; i < 8; ++i) {
        int r = i + 8 * hi;
        stage[wv][1][r][mrow]      = m0[i];
        stage[wv][1][r][mrow + 16] = m1[i];
    }
    __builtin_amdgcn_wave_barrier();
    v2f am2[8];
#pragma unroll
    for (int ks = 0; ks < 8; ++ks) {
        int k0 = 4 * ks + 2 * hi;
        am2[ks].x = stage[wv][1][mrow][k0];
        am2[ks].y = stage[wv][1][mrow][k0 + 1];
    }
    v8f gx[6], gh[6];
#pragma unroll
    for (int nt = 0; nt < 6; ++nt) {
        v8f ax = z8(), ah = z8();
        int n = nt * 16 + mrow;
#pragma unroll
        for (int ks = 0; ks < 8; ++ks) {
            int k0 = 4 * ks + 2 * hi;
            v2f bi, bh;
            bi.x = w_ih[n * 32 + k0]; bi.y = w_ih[n * 32 + k0 + 1];
            bh.x = w_hh[n * 32 + k0]; bh.y = w_hh[n * 32 + k0 + 1];
            ax = wmma4(am2[ks], bi, ax);
            ah = wmma4(aht[ks], bh, ah);
        }
        gx[nt] = ax;
        gh[nt] = ah;
    }

    // ---- gates + h_next store ----
#pragma unroll
    for (int j = 0; j < 2; ++j) {
        int n = j * 16 + mrow;  // hd offset within H
        float bir = b_ih[n],      bhr = b_hh[n];
        float biz = b_ih[32 + n], bhz = b_hh[32 + n];
        float bin = b_ih[64 + n], bhn = b_hh[64 + n];
#pragma unroll
        for (int i = 0; i < 8; ++i) {
            int r = i + 8 * hi;
            int c = cbase + r;
            int cc2 = c < C_ ? c : C_ - 1;
            float rr = sigm(gx[j][i] + bir + gh[j][i] + bhr);
            float zz = sigm(gx[2 + j][i] + biz + gh[2 + j][i] + bhz);
            float nn = tanhf(gx[4 + j][i] + bin + rr * (gh[4 + j][i] + bhn));
            float hp = ht_in[((size_t)b * C_ + cc2) * H_ + n];
            float hn = (1.f - zz) * nn + zz * hp;
            if (c < C_) ht_out[((size_t)b * C_ + c) * H_ + n] = hn;
        }
    }
}

// pred[b,t] = sigmoid(h_next[b, questions[b,t+1], :] . pred_w + pred_b)
__global__ __launch_bounds__(32) void gkt_pred(const int* __restrict__ questions,
                                               const float* __restrict__ ht_next,
                                               const float* __restrict__ pred_w,
                                               const float* __restrict__ pred_b,
                                               float* __restrict__ out, int t)
{
    int b = blockIdx.x;
    int lane = threadIdx.x;
    int qn = questions[b * T_ + (t + 1)];
    float v = ht_next[((size_t)b * C_ + qn) * H_ + lane] * pred_w[lane];
#pragma unroll
    for (int off = 16; off > 0; off >>= 1) v += __shfl_xor(v, off, 32);
    if (lane == 0) out[b * (T_ - 1) + t] = 1.f / (1.f + __expf(-(v + pred_b[0])));
}

extern "C" void kernel_launch(void* const* d_in, const int* in_sizes, int n_in,
                              void* d_out, int out_size, void* d_ws, size_t ws_size,
                              hipStream_t stream)
{
    (void)in_sizes; (void)n_in; (void)out_size; (void)ws_size;
    const int*   features  = (const int*)d_in[0];
    const int*   questions = (const int*)d_in[1];
    const float* graph     = (const float*)d_in[2];
    const float* emb_x     = (const float*)d_in[3];
    const float* emb_c     = (const float*)d_in[4];
    const float* self_w1   = (const float*)d_in[5];
    const float* self_b1   = (const float*)d_in[6];
    const float* self_w2   = (const float*)d_in[7];
    const float* self_b2   = (const float*)d_in[8];
    const float* n0_w1     = (const float*)d_in[9];
    const float* n0_b1     = (const float*)d_in[10];
    const float* n0_w2     = (const float*)d_in[11];
    const float* n0_b2     = (const float*)d_in[12];
    const float* n1_w1     = (const float*)d_in[13];
    const float* n1_b1     = (const float*)d_in[14];
    const float* n1_w2     = (const float*)d_in[15];
    const float* n1_b2     = (const float*)d_in[16];
    const float* ea_w      = (const float*)d_in[17];
    const float* ea_ew     = (const float*)d_in[18];
    const float* ea_eb     = (const float*)d_in[19];
    const float* ea_aw     = (const float*)d_in[20];
    const float* ea_ab     = (const float*)d_in[21];
    const float* gru_w_ih  = (const float*)d_in[22];
    const float* gru_w_hh  = (const float*)d_in[23];
    const float* gru_b_ih  = (const float*)d_in[24];
    const float* gru_b_hh  = (const float*)d_in[25];
    const float* pred_w    = (const float*)d_in[26];
    const float* pred_b    = (const float*)d_in[27];

    size_t hbytes = (size_t)B_ * C_ * H_ * sizeof(float);
    size_t halign = ((hbytes + 255) / 256) * 256;
    float* h0 = (float*)d_ws;
    float* h1 = (float*)((char*)d_ws + halign);

    hipMemsetAsync(h0, 0, hbytes, stream);  // ht0 = zeros (capturable)

    for (int t = 0; t < T_ - 1; ++t) {
        const float* hin = (t & 1) ? h1 : h0;
        float*       hout = (t & 1) ? h0 : h1;
        gkt_step<<<B_ * 8, 128, 0, stream>>>(
            features, questions, graph, emb_x, emb_c,
            self_w1, self_b1, self_w2, self_b2,
            n0_w1, n0_b1, n0_w2, n0_b2,
            n1_w1, n1_b1, n1_w2, n1_b2,
            ea_w, ea_ew, ea_eb, ea_aw, ea_ab,
            gru_w_ih, gru_w_hh, gru_b_ih, gru_b_hh,
            hin, hout, t);
        gkt_pred<<<B_, 32, 0, stream>>>(questions, hout, pred_w, pred_b,
                                        (float*)d_out, t);
    }
}